// NeighborlistBruteNsq_66486093742352
// MI455X (gfx1250) — compile-verified
//
#include <hip/hip_runtime.h>
#include <hip/hip_bf16.h>

#define NATOMS 4096
#define TPB    256
#define RCUT   0.5f

typedef float vfloat4 __attribute__((ext_vector_type(4)));

#define AS1 __attribute__((address_space(1)))
#define AS3 __attribute__((address_space(3)))

// ---- CDNA5 async global->LDS copy (ASYNCcnt-tracked DMA, gfx1250) ----------
// One 32-bit element per enabled lane, copied from global address g+OFF to LDS address l+OFF.
// OFF is a template parameter so it stays an ICE for the builtin's immediate
// offset field (lets the three dword asyncs share one 64-bit address VGPR
// pair and clause together with offset:0/4/8).
template <int OFF>
__device__ __forceinline__ void async_copy_b32(const void* gsrc, void* lds_generic) {
    unsigned long long g = (unsigned long long)gsrc;
    // Low 32 bits of a generic pointer into the LDS aperture are the LDS byte
    // offset (flat->LDS translation truncates to addr[31:0]).
    unsigned l = (unsigned)(unsigned long long)lds_generic;
#if __has_builtin(__builtin_amdgcn_global_load_async_to_lds_b32)
    __builtin_amdgcn_global_load_async_to_lds_b32(
        (AS1 int*)g, (AS3 int*)(unsigned long long)l, OFF, /*cpol*/0);
#else
    asm volatile("global_load_async_to_lds_b32 %0, %1, off offset:%c2"
                 :: "v"(l), "v"(g), "i"(OFF) : "memory");
#endif
}

__device__ __forceinline__ void wait_async0() {
#if __has_builtin(__builtin_amdgcn_s_wait_asynccnt)
    __builtin_amdgcn_s_wait_asynccnt(0);
#else
    asm volatile("s_wait_asynccnt 0x0" ::: "memory");
#endif
}

// Grid: (NATOMS/TPB, NATOMS). blockIdx.y = row atom i, x-blocks tile j.
// Pair (i, j) with j > i maps to triu row-major slot  i*(2N-1-i)/2 + (j-i-1),
// exactly np.triu_indices(N, k=1) order, so i_pairs/j_pairs are never read
// (saves 67 MB of the ~200 MB naive traffic; kernel is store-BW bound:
// 134 MB of NT B128 stores ~ 5.8 us at 23.3 TB/s).
__global__ __launch_bounds__(TPB)
void nbl_pairs_kernel(const float* __restrict__ pos,   // [N,3]
                      const float* __restrict__ boxv,  // [3,3]
                      float* __restrict__ out)         // [P,4]
{
    __shared__ float s_pos[TPB * 3 + 4]; // [0..767]: chunk j positions, [768..770]: pos_i

    const int i     = (int)blockIdx.y;
    const int jbase = (int)blockIdx.x * TPB;
    const int tid   = (int)threadIdx.x;

    // Entire block below/on the diagonal -> no valid pairs.
    if (jbase + TPB - 1 <= i) return;

    const int j = jbase + tid;             // always < NATOMS (grid covers N exactly)

    // ---- async-stage this block's positions into LDS -----------------------
    {
        const float* gj = pos + (size_t)j * 3;     // this lane's j atom (3 dwords)
        float* lj = &s_pos[tid * 3];
        async_copy_b32<0>(gj, lj);
        async_copy_b32<4>(gj, lj);
        async_copy_b32<8>(gj, lj);
        if (tid < 3) {                             // broadcast row atom i
            async_copy_b32<0>(pos + (size_t)i * 3 + tid, &s_pos[TPB * 3 + tid]);
        }
    }
    wait_async0();        // ASYNCcnt -> 0: DMA'd LDS data visible to this wave
    __syncthreads();      // publish lane0-2's pos_i staging to all waves

    // Orthogonal box lengths (diagonal); uniform scalar loads.
    const float bx = boxv[0], by = boxv[4], bz = boxv[8];
    const float hx = 0.5f * bx, hy = 0.5f * by, hz = 0.5f * bz;

    if (j > i) {
        const float pix = s_pos[TPB * 3 + 0];
        const float piy = s_pos[TPB * 3 + 1];
        const float piz = s_pos[TPB * 3 + 2];
        float dx = pix - s_pos[tid * 3 + 0];
        float dy = piy - s_pos[tid * 3 + 1];
        float dz = piz - s_pos[tid * 3 + 2];

        // jnp.remainder(r + half, box) - half  (floor-form, sign of divisor)
        float wx = dx + hx; wx -= floorf(wx / bx) * bx; dx = wx - hx;
        float wy = dy + hy; wy -= floorf(wy / by) * by; dy = wy - hy;
        float wz = dz + hz; wz -= floorf(wz / bz) * bz; dz = wz - hz;

        const float d = sqrtf(dx * dx + dy * dy + dz * dz);
        const float m = (d <= RCUT) ? 1.0f : 0.0f;

        const unsigned long long base =
            ((unsigned long long)i * (unsigned long long)(2 * NATOMS - 1 - i)) >> 1;
        const unsigned long long idx = base + (unsigned long long)(j - i - 1);

        vfloat4 r;
        r.x = dx * m; r.y = dy * m; r.z = dz * m; r.w = d * m;
        // streaming 16B store (B128, non-temporal): 134 MB write-once output
        __builtin_nontemporal_store(r, (vfloat4*)out + idx);
    }
}

extern "C" void kernel_launch(void* const* d_in, const int* in_sizes, int n_in,
                              void* d_out, int out_size, void* d_ws, size_t ws_size,
                              hipStream_t stream) {
    (void)in_sizes; (void)n_in; (void)out_size; (void)d_ws; (void)ws_size;
    const float* pos  = (const float*)d_in[0];  // positions [4096,3] f32
    const float* boxv = (const float*)d_in[1];  // box_vectors [3,3] f32
    // d_in[2]/d_in[3] (i_pairs/j_pairs) intentionally unread: canonical
    // row-major triu order is recomputed analytically in-kernel.
    dim3 grid(NATOMS / TPB, NATOMS, 1);
    nbl_pairs_kernel<<<grid, dim3(TPB, 1, 1), 0, stream>>>(pos, boxv, (float*)d_out);
}